// Interp2_68719477102
// MI455X (gfx1250) — compile-verified
//
#include <hip/hip_runtime.h>
#include <stdint.h>

// Problem constants (fixed by the reference).
constexpr int Bc  = 8;
constexpr int Cc  = 16;
constexpr int Hc  = 512;
constexpr int Wc  = 512;
constexpr int HWc = Hc * Wc;              // 262144 = 2^18

// ---------------------------------------------------------------------------
// Kernel 1: BCHW -> BHWC layout transform for a group of batches.
// Uses the CDNA5 async global->LDS path: each lane issues
// GLOBAL_LOAD_ASYNC_TO_LDS_B32 with a per-lane LDS destination, fusing the
// coalesced global read and the transposing LDS scatter into one instruction
// (no VGPR staging). Completion is tracked on ASYNCcnt; we drain it with
// s_wait_asynccnt 0 before the workgroup barrier. LDS rows padded to stride
// 17 (gcd(17,64)=1) -> conflict-free on the 64-bank LDS for both the scatter
// and the coalesced read-out. v is read exactly once -> TH_LOAD_NT so the
// stream doesn't evict the L2-resident BHWC copy.
// ---------------------------------------------------------------------------
__global__ void tr_bchw2bhwc(const float* __restrict__ v,
                             float* __restrict__ vt, int b0) {
    __shared__ float lds[64 * 17];
    const int tile = blockIdx.x;
    const int bb   = tile >> 12;                 // tile / (HW/64)
    const int hw0  = (tile & 4095) << 6;         // (tile % 4096) * 64
    const int b    = b0 + bb;
    const int tid  = threadIdx.x;
    const int hwl  = tid & 63;
    const int cb   = tid >> 6;                   // 0..3

    const float* src = v + (size_t)b * Cc * HWc + hw0 + hwl;
#pragma unroll
    for (int k = 0; k < 4; ++k) {
        const int c = cb + (k << 2);             // covers c = 0..15
        // LDS byte offset of lds[hwl*17 + c] (flat->LDS aperture: low 32 bits)
        const uint32_t ldsaddr = (uint32_t)(uintptr_t)(&lds[hwl * 17 + c]);
        const uint64_t gaddr   = (uint64_t)(uintptr_t)(src + (size_t)c * HWc);
        asm volatile("global_load_async_to_lds_b32 %0, %1, off th:TH_LOAD_NT"
                     :: "v"(ldsaddr), "v"(gaddr) : "memory");
    }
    // Drain this wave's async global->LDS transfers, then rendezvous so every
    // wave's scatter is visible before the read-out phase.
    asm volatile("s_wait_asynccnt 0x0" ::: "memory");
    __syncthreads();

    float* dst = vt + ((size_t)bb * HWc + hw0) * Cc;
#pragma unroll
    for (int k = 0; k < 4; ++k) {
        const int e = tid + (k << 8);            // consecutive lanes -> consecutive addrs
        dst[e] = lds[(e >> 4) * 17 + (e & 15)];
    }
}

// ---------------------------------------------------------------------------
// Kernel 2: bilinear gather from the BHWC copy.
// One thread per query (b,h,w). Each corner is one contiguous 64B read
// (4x global_load_b128) -> exactly the useful bytes, served from L2 since
// the BHWC copy fits in the 192MB L2. Output written per-channel with hw
// varying across lanes -> coalesced 128B non-temporal stores.
// ---------------------------------------------------------------------------
__global__ void bilerp_bhwc(const float* __restrict__ vt,
                            const float* __restrict__ xq,
                            const float* __restrict__ yq,
                            float* __restrict__ out, int b0, int g) {
    const int q = blockIdx.x * blockDim.x + threadIdx.x;
    if (q >= g * HWc) return;
    const int bb = q >> 18;                      // q / HWc
    const int hw = q & (HWc - 1);
    const int b  = b0 + bb;

    const float x = __builtin_nontemporal_load(xq + (size_t)b * HWc + hw);
    const float y = __builtin_nontemporal_load(yq + (size_t)b * HWc + hw);

    const bool inv = (x < 0.0f) || (x >= (float)Wc) || (y < 0.0f) || (y >= (float)Hc);

    int x0 = (int)floorf(x); x0 = x0 < 0 ? 0 : (x0 > Wc - 1 ? Wc - 1 : x0);
    int y0 = (int)floorf(y); y0 = y0 < 0 ? 0 : (y0 > Hc - 1 ? Hc - 1 : y0);
    const int x1 = (x0 + 1 > Wc - 1) ? Wc - 1 : x0 + 1;
    const int y1 = (y0 + 1 > Hc - 1) ? Hc - 1 : y0 + 1;

    const float dx = x - (float)x0;
    const float dy = y - (float)y0;
    float w00 = (1.0f - dy) * (1.0f - dx);
    float w01 = (1.0f - dy) * dx;
    float w10 = dy * (1.0f - dx);
    float w11 = dy * dx;
    if (inv) { w00 = w01 = w10 = w11 = 0.0f; }   // exact zero, matches reference mask

    const float4* p00 = (const float4*)(vt + (((size_t)bb * Hc + y0) * Wc + x0) * (size_t)Cc);
    const float4* p01 = (const float4*)(vt + (((size_t)bb * Hc + y0) * Wc + x1) * (size_t)Cc);
    const float4* p10 = (const float4*)(vt + (((size_t)bb * Hc + y1) * Wc + x0) * (size_t)Cc);
    const float4* p11 = (const float4*)(vt + (((size_t)bb * Hc + y1) * Wc + x1) * (size_t)Cc);

    float acc[16];
#pragma unroll
    for (int k = 0; k < 4; ++k) {
        const float4 a = p00[k], bq = p01[k], cq = p10[k], dq = p11[k];
        acc[4 * k + 0] = a.x * w00 + bq.x * w01 + cq.x * w10 + dq.x * w11;
        acc[4 * k + 1] = a.y * w00 + bq.y * w01 + cq.y * w10 + dq.y * w11;
        acc[4 * k + 2] = a.z * w00 + bq.z * w01 + cq.z * w10 + dq.z * w11;
        acc[4 * k + 3] = a.w * w00 + bq.w * w01 + cq.w * w10 + dq.w * w11;
    }

    float* o = out + (size_t)b * Cc * HWc + hw;
#pragma unroll
    for (int c = 0; c < Cc; ++c)
        __builtin_nontemporal_store(acc[c], o + (size_t)c * HWc);
}

// ---------------------------------------------------------------------------
// Fallback: direct gather from BCHW (only if workspace < 16 MiB).
// ---------------------------------------------------------------------------
__global__ void bilerp_bchw(const float* __restrict__ v,
                            const float* __restrict__ xq,
                            const float* __restrict__ yq,
                            float* __restrict__ out) {
    const int q = blockIdx.x * blockDim.x + threadIdx.x;
    if (q >= Bc * HWc) return;
    const int b  = q >> 18;
    const int hw = q & (HWc - 1);

    const float x = xq[(size_t)b * HWc + hw];
    const float y = yq[(size_t)b * HWc + hw];
    const bool inv = (x < 0.0f) || (x >= (float)Wc) || (y < 0.0f) || (y >= (float)Hc);

    int x0 = (int)floorf(x); x0 = x0 < 0 ? 0 : (x0 > Wc - 1 ? Wc - 1 : x0);
    int y0 = (int)floorf(y); y0 = y0 < 0 ? 0 : (y0 > Hc - 1 ? Hc - 1 : y0);
    const int x1 = (x0 + 1 > Wc - 1) ? Wc - 1 : x0 + 1;
    const int y1 = (y0 + 1 > Hc - 1) ? Hc - 1 : y0 + 1;

    const float dx = x - (float)x0;
    const float dy = y - (float)y0;
    float w00 = (1.0f - dy) * (1.0f - dx);
    float w01 = (1.0f - dy) * dx;
    float w10 = dy * (1.0f - dx);
    float w11 = dy * dx;
    if (inv) { w00 = w01 = w10 = w11 = 0.0f; }

    const float* vb = v + (size_t)b * Cc * HWc;
    float* o = out + (size_t)b * Cc * HWc + hw;
#pragma unroll
    for (int c = 0; c < Cc; ++c) {
        const float* vc = vb + (size_t)c * HWc;
        const float v00 = vc[y0 * Wc + x0];
        const float v01 = vc[y0 * Wc + x1];
        const float v10 = vc[y1 * Wc + x0];
        const float v11 = vc[y1 * Wc + x1];
        __builtin_nontemporal_store(v00 * w00 + v01 * w01 + v10 * w10 + v11 * w11,
                                    o + (size_t)c * HWc);
    }
}

extern "C" void kernel_launch(void* const* d_in, const int* in_sizes, int n_in,
                              void* d_out, int out_size, void* d_ws, size_t ws_size,
                              hipStream_t stream) {
    const float* v  = (const float*)d_in[0];
    const float* xq = (const float*)d_in[1];
    const float* yq = (const float*)d_in[2];
    float* out      = (float*)d_out;

    const size_t perBatchBytes = (size_t)HWc * Cc * sizeof(float);   // 16 MiB
    size_t gmax = ws_size / perBatchBytes;
    if (gmax > (size_t)Bc) gmax = Bc;

    if (gmax >= 1) {
        float* vt = (float*)d_ws;
        for (int b0 = 0; b0 < Bc; b0 += (int)gmax) {
            const int g = ((int)gmax < Bc - b0) ? (int)gmax : (Bc - b0);
            // layout transform for batches [b0, b0+g)
            tr_bchw2bhwc<<<dim3(g * (HWc / 64)), dim3(256), 0, stream>>>(v, vt, b0);
            // gather for the same batches (stream order guarantees the dependency)
            const int total  = g * HWc;
            const int blocks = (total + 255) / 256;
            bilerp_bhwc<<<dim3(blocks), dim3(256), 0, stream>>>(vt, xq, yq, out, b0, g);
        }
    } else {
        const int total  = Bc * HWc;
        const int blocks = (total + 255) / 256;
        bilerp_bchw<<<dim3(blocks), dim3(256), 0, stream>>>(v, xq, yq, out);
    }
}